// MoERouter_56032143343805
// MI455X (gfx1250) — compile-verified
//
#include <hip/hip_runtime.h>

typedef __attribute__((ext_vector_type(2))) float v2f;
typedef __attribute__((ext_vector_type(8))) float v8f;

#define N_TOK   32768
#define DIM     1024
#define NEXP    64
#define CAP     768
#define KC      64          // K-panel width staged through LDS
#define LDSPAD  2           // row pad -> conflict-free ds_load_b64 fragment reads

// ---------------------------------------------------------------------------
// Kernel 1: router GEMM (f32 WMMA 16x16x4) + softmax + top-2
// 1 wave = 16 tokens x 64 experts (4 accumulator tiles).
// ---------------------------------------------------------------------------
__device__ __forceinline__ void top2_insert(float v, int i,
                                            float& m1, int& i1,
                                            float& m2, int& i2) {
    bool b1 = (v > m1) || (v == m1 && i < i1);
    if (b1) { m2 = m1; i2 = i1; m1 = v; i1 = i; }
    else {
        bool b2 = (v > m2) || (v == m2 && i < i2);
        if (b2) { m2 = v; i2 = i; }
    }
}

__global__ __launch_bounds__(128)
void router_kernel(const float* __restrict__ x,
                   const float* __restrict__ W,
                   float* __restrict__ router_probs,   // [N,64]
                   float* __restrict__ topk_probs,     // [N,2] (normalized)
                   int*   __restrict__ topk_idx) {     // [N,2]
    __shared__ float Wls[NEXP][KC + LDSPAD];
    __shared__ float xls[4][16][KC + LDSPAD];

    const int tid  = threadIdx.x;
    const int wave = tid >> 5;
    const int lane = tid & 31;
    const int half = lane >> 4;      // 0: K-lo / tokens r, 1: K-hi / tokens r+8
    const int l16  = lane & 15;
    const int tile = blockIdx.x * 4 + wave;   // 0..2047
    const int tok0 = tile * 16;

    v8f acc0 = {}, acc1 = {}, acc2 = {}, acc3 = {};

    for (int kp = 0; kp < DIM; kp += KC) {
        // --- cooperative, coalesced panel loads (float4 from global) ---
        // W panel: 64 rows x 64 cols = 1024 float4; 128 threads x 8
        #pragma unroll
        for (int i = 0; i < 2; ++i) {
            int c    = tid + i * 128;        // 0..255
            int row  = c >> 4;               // 0..15  (x4 rows below)
            int col4 = c & 15;
            // rows 0..63: c>>4 gives 0..15, need 4 passes -> fold into i loop
        }
        // (re-do properly: 4096 floats = 1024 float4 chunks, 128 thr x 8)
        #pragma unroll
        for (int i = 0; i < 8; ++i) {
            int c    = tid + i * 128;        // 0..1023
            int row  = c >> 4;               // 0..63
            int col4 = c & 15;               // 0..15
            float4 wv = *(const float4*)(W + row * DIM + kp + col4 * 4);
            float* dst = &Wls[row][col4 * 4];
            dst[0] = wv.x; dst[1] = wv.y; dst[2] = wv.z; dst[3] = wv.w;
        }
        // x panel: this wave's 16 token rows x 64 cols = 256 float4; 32 ln x 8
        #pragma unroll
        for (int i = 0; i < 8; ++i) {
            int c    = lane + i * 32;        // 0..255
            int row  = c >> 4;               // 0..15
            int col4 = c & 15;
            float4 xv = *(const float4*)(x + (size_t)(tok0 + row) * DIM + kp + col4 * 4);
            float* dst = &xls[wave][row][col4 * 4];
            dst[0] = xv.x; dst[1] = xv.y; dst[2] = xv.z; dst[3] = xv.w;
        }
        if (kp + KC < DIM)  // hint next panel into cache (global_prefetch_b8)
            __builtin_prefetch(x + (size_t)(tok0 + l16) * DIM + kp + KC, 0, 1);
        __syncthreads();

        // --- WMMA over the panel: A=16x4 tokens, B=4x16 experts, 4 e-tiles ---
        #pragma unroll
        for (int k = 0; k < KC; k += 4) {
            const int ko = k + 2 * half;     // even -> 8B aligned ds_load_b64
            v2f a  = *(const v2f*)&xls[wave][l16][ko];
            v2f b0 = *(const v2f*)&Wls[ 0 + l16][ko];
            v2f b1 = *(const v2f*)&Wls[16 + l16][ko];
            v2f b2 = *(const v2f*)&Wls[32 + l16][ko];
            v2f b3 = *(const v2f*)&Wls[48 + l16][ko];
            acc0 = __builtin_amdgcn_wmma_f32_16x16x4_f32(false, a, false, b0, (short)0, acc0, false, false);
            acc1 = __builtin_amdgcn_wmma_f32_16x16x4_f32(false, a, false, b1, (short)0, acc1, false, false);
            acc2 = __builtin_amdgcn_wmma_f32_16x16x4_f32(false, a, false, b2, (short)0, acc2, false, false);
            acc3 = __builtin_amdgcn_wmma_f32_16x16x4_f32(false, a, false, b3, (short)0, acc3, false, false);
        }
        __syncthreads();
    }

    // --- epilogue: per-token softmax over 64 experts + top-2 ---
    // C layout: VGPR r, half h -> token tok0 + r + 8h ; lane%16 -> expert%16.
    #pragma unroll
    for (int r = 0; r < 8; ++r) {
        const int token = tok0 + r + 8 * half;
        float v0 = acc0[r], v1 = acc1[r], v2 = acc2[r], v3 = acc3[r];

        float mx = fmaxf(fmaxf(v0, v1), fmaxf(v2, v3));
        #pragma unroll
        for (int s = 1; s <= 8; s <<= 1) mx = fmaxf(mx, __shfl_xor(mx, s, 32));

        float e0 = __expf(v0 - mx), e1 = __expf(v1 - mx);
        float e2 = __expf(v2 - mx), e3 = __expf(v3 - mx);
        float sum = e0 + e1 + e2 + e3;
        #pragma unroll
        for (int s = 1; s <= 8; s <<= 1) sum += __shfl_xor(sum, s, 32);
        float inv = __frcp_rn(sum);
        float p0 = e0 * inv, p1 = e1 * inv, p2 = e2 * inv, p3 = e3 * inv;

        float* rp = router_probs + (size_t)token * NEXP;
        rp[ 0 + l16] = p0; rp[16 + l16] = p1; rp[32 + l16] = p2; rp[48 + l16] = p3;

        float m1 = p0; int i1 = l16;
        float m2 = -1.0f; int i2 = -1;
        top2_insert(p1, 16 + l16, m1, i1, m2, i2);
        top2_insert(p2, 32 + l16, m1, i1, m2, i2);
        top2_insert(p3, 48 + l16, m1, i1, m2, i2);
        #pragma unroll
        for (int s = 1; s <= 8; s <<= 1) {
            float om1 = __shfl_xor(m1, s, 32); int oi1 = __shfl_xor(i1, s, 32);
            float om2 = __shfl_xor(m2, s, 32); int oi2 = __shfl_xor(i2, s, 32);
            top2_insert(om1, oi1, m1, i1, m2, i2);
            top2_insert(om2, oi2, m1, i1, m2, i2);
        }
        if (l16 == 0) {
            float ti = __frcp_rn(m1 + m2);
            topk_probs[token * 2 + 0] = m1 * ti;
            topk_probs[token * 2 + 1] = m2 * ti;
            topk_idx[token * 2 + 0]   = i1;
            topk_idx[token * 2 + 1]   = i2;
        }
    }
}

// ---------------------------------------------------------------------------
// Kernel 2: per-chunk expert histograms (chunk = 1024 flat slots)
// ---------------------------------------------------------------------------
__global__ __launch_bounds__(256)
void hist_kernel(const int* __restrict__ idx, int* __restrict__ hist) {
    __shared__ int h[NEXP];
    int tid = threadIdx.x;
    if (tid < NEXP) h[tid] = 0;
    __syncthreads();
    int base = blockIdx.x * 1024;
    #pragma unroll
    for (int i = 0; i < 4; ++i)
        atomicAdd(&h[idx[base + tid + i * 256]], 1);
    __syncthreads();
    if (tid < NEXP) hist[blockIdx.x * NEXP + tid] = h[tid];
}

// ---------------------------------------------------------------------------
// Kernel 3: exclusive scan of chunk histograms per expert (64 chunks)
// ---------------------------------------------------------------------------
__global__ __launch_bounds__(64)
void scan_kernel(const int* __restrict__ hist, int* __restrict__ offs) {
    int e = threadIdx.x;
    int run = 0;
    for (int c = 0; c < 64; ++c) {
        offs[c * NEXP + e] = run;
        run += hist[c * NEXP + e];
    }
}

// ---------------------------------------------------------------------------
// Kernel 4: stable within-chunk position via wave32 ballot match-any -> keep
// ---------------------------------------------------------------------------
__global__ __launch_bounds__(32)
void keep_kernel(const int* __restrict__ idx, const int* __restrict__ offs,
                 int* __restrict__ keep) {
    __shared__ int cnt[NEXP];
    int lane = threadIdx.x;
    cnt[lane]      = offs[blockIdx.x * NEXP + lane];
    cnt[lane + 32] = offs[blockIdx.x * NEXP + lane + 32];
    __syncthreads();
    int base = blockIdx.x * 1024;
    for (int g = 0; g < 32; ++g) {
        int i = base + g * 32 + lane;
        int e = idx[i];
        unsigned mask = 0xFFFFFFFFu;
        #pragma unroll
        for (int b = 0; b < 6; ++b) {
            unsigned v = (unsigned)__ballot((e >> b) & 1);
            mask &= ((e >> b) & 1) ? v : ~v;
        }
        unsigned lower = mask & ((1u << lane) - 1u);
        int pos = cnt[e] + __popc(lower);
        keep[i] = (pos < CAP) ? 1 : 0;
        __syncthreads();
        if (lower == 0u) atomicAdd(&cnt[e], __popc(mask));  // first occurrence updates
        __syncthreads();
    }
}

// ---------------------------------------------------------------------------
// Kernel 5: dense dispatch/combine write ([N,64,2], coalesced float2)
// ---------------------------------------------------------------------------
__global__ __launch_bounds__(256)
void out_kernel(const int* __restrict__ idx, const int* __restrict__ keep,
                const float* __restrict__ tkp,
                float* __restrict__ dispatch, float* __restrict__ combine) {
    int t = blockIdx.x * 256 + threadIdx.x;   // 0 .. N*64-1
    int n = t >> 6;
    int e = t & 63;
    int i0 = idx[2 * n],  i1 = idx[2 * n + 1];
    int k0 = keep[2 * n], k1 = keep[2 * n + 1];
    float d0 = (e == i0 && k0) ? 1.0f : 0.0f;
    float d1 = (e == i1 && k1) ? 1.0f : 0.0f;
    float2 dv = make_float2(d0, d1);
    float2 cv = make_float2(d0 * tkp[2 * n], d1 * tkp[2 * n + 1]);
    *(float2*)(dispatch + (size_t)t * 2) = dv;
    *(float2*)(combine  + (size_t)t * 2) = cv;
}

// ---------------------------------------------------------------------------
extern "C" void kernel_launch(void* const* d_in, const int* in_sizes, int n_in,
                              void* d_out, int out_size, void* d_ws, size_t ws_size,
                              hipStream_t stream) {
    const float* x = (const float*)d_in[0];
    const float* W = (const float*)d_in[1];

    float* out      = (float*)d_out;
    float* dispatch = out;                         // [N,64,2] = 4194304
    float* combine  = out + 4194304;               // [N,64,2]
    float* rprobs   = out + 8388608;               // [N,64]   = 2097152
    float* tkp      = out + 10485760;              // [N,2]    = 65536

    char* ws  = (char*)d_ws;
    int* idx  = (int*)(ws);                        // 65536 ints
    int* hist = (int*)(ws + 262144);               // 4096 ints
    int* offs = (int*)(ws + 262144 + 16384);       // 4096 ints
    int* keep = (int*)(ws + 262144 + 32768);       // 65536 ints

    router_kernel<<<512, 128, 0, stream>>>(x, W, rprobs, tkp, idx);
    hist_kernel<<<64, 256, 0, stream>>>(idx, hist);
    scan_kernel<<<1, 64, 0, stream>>>(hist, offs);
    keep_kernel<<<64, 32, 0, stream>>>(idx, offs, keep);
    out_kernel<<<8192, 256, 0, stream>>>(idx, keep, tkp, dispatch, combine);
}